// moeStack_31275951850277
// MI455X (gfx1250) — compile-verified
//
#include <hip/hip_runtime.h>

typedef __attribute__((ext_vector_type(2))) float v2f;
typedef __attribute__((ext_vector_type(8))) float v8f;

#define BB 64   // batch
#define EE 20   // experts
#define KK 3    // top-k
#define HHD 3   // heads
#define HE 60   // H*E

// ---------------------------------------------------------------------------
// Generic WMMA-f32 linear:  Y[M,N] = act( X[M,Kd] @ W[Kd,N] + b[N] )
//
// 4 waves / block; each wave owns one 16x16 output tile; all 4 waves share the
// same 16 M-rows, so the A tile (16 x 64, pitch 68 to avoid bank conflicts) is
// staged in LDS once per K-chunk and read back with ds_load_2addr_b64.
// W's leading dimension is a template parameter so all 32 B-operand loads per
// K-chunk use compile-time immediate offsets off one base pointer (no per-step
// 64-bit address arithmetic).
// Out-of-range rows/cols are handled by *address clamping* (garbage values only
// ever land in D elements we never store), so the K loop is branch-free and
// EXEC is all-ones at every v_wmma_f32_16x16x4_f32.
// Requires: Kd % 64 == 0, ldx % 4 == 0 (true for every call site here).
// ---------------------------------------------------------------------------
template <int LDW>
__global__ __launch_bounds__(128) void wmma_linear_kernel(
    const float* __restrict__ X, long xB, int ldx,
    const float* __restrict__ W, long wB,
    const float* __restrict__ Bv, long bB,
    float* __restrict__ Y, long yB, int ldy,
    int M, int N, int Kd, int act)
{
    __shared__ float As[16][68];   // 16 rows x 64 K, pitch 68 (=4 mod 64)

    const int wave = threadIdx.x >> 5;
    const int lane = threadIdx.x & 31;
    const int half = lane >> 4;      // 0: K=0,1   1: K=2,3
    const int l16  = lane & 15;
    const int nBase = (blockIdx.x * 4 + wave) * 16;
    const int mBase = blockIdx.y * 16;
    const int e = blockIdx.z;

    X  += (long)e * xB;
    W  += (long)e * wB;
    Bv += (long)e * bB;
    Y  += (long)e * yB;

    const int colB = nBase + l16;
    const int colC = (colB < N) ? colB : (N - 1);       // clamped, always legal

    // Cooperative A-tile loader mapping: 16 rows x 8 threads x 8 floats.
    const int ldRow  = threadIdx.x >> 3;                // 0..15
    const int ldCol  = (threadIdx.x & 7) * 8;           // 0,8,..,56
    int gRow = mBase + ldRow;
    gRow = (gRow < M) ? gRow : (M - 1);                 // clamped, always legal
    const float* __restrict__ xTile = X + (long)gRow * ldx + ldCol;

    v8f c = {0.f, 0.f, 0.f, 0.f, 0.f, 0.f, 0.f, 0.f};

    for (int k0 = 0; k0 < Kd; k0 += 64) {
        // ---- stage A[mBase:mBase+16, k0:k0+64] into LDS ----
        {
            const float4* src = reinterpret_cast<const float4*>(xTile + k0);
            const float4 t0 = src[0];
            const float4 t1 = src[1];
            float4* dst = reinterpret_cast<float4*>(&As[ldRow][ldCol]);
            dst[0] = t0;
            dst[1] = t1;
        }
        // prefetch next W chunk for this lane's column (branch-free clamp)
        {
            const int kpf = (k0 + 64 < Kd) ? (k0 + 64) : (Kd - 1);
            __builtin_prefetch(W + (long)kpf * LDW + colC, 0, 0);
        }
        __syncthreads();

        // ---- 16 WMMA steps over this K-chunk ----
        // Single base address; every load below has a compile-time offset.
        const float* __restrict__ wP = W + (long)(k0 + half * 2) * LDW + colC;
#pragma unroll
        for (int kc = 0; kc < 64; kc += 4) {
            v2f a, b;
            const float2 ta =
                *reinterpret_cast<const float2*>(&As[l16][kc + half * 2]); // ds_load b64
            a.x = ta.x; a.y = ta.y;
            b.x = wP[(long)kc * LDW];            // imm-offset global_load_b32
            b.y = wP[(long)kc * LDW + LDW];      // imm-offset global_load_b32
            c = __builtin_amdgcn_wmma_f32_16x16x4_f32(
                    false, a, false, b, (short)0, c, false, false);
        }
        __syncthreads();
    }

    // ---- epilogue: bias + activation + guarded store ----
    if (colB < N) {
        const float bias = Bv[colB];
#pragma unroll
        for (int r = 0; r < 8; ++r) {
            const int row = mBase + r + half * 8;
            if (row < M) {
                float y = c[r] + bias;
                if (act == 1)      y = fmaxf(y, 0.f);
                else if (act == 2) y = 1.f / (1.f + __expf(-y));
                Y[(long)row * ldy + colB] = y;
            }
        }
    }
}

// ---------------------------------------------------------------------------
// Fused attention -> softmax(axis=j) -> sum(axis=i) gate scores.
// One workgroup per (b, e). q/k: (B, S, H, E). Output gp: (B, E, S).
// ---------------------------------------------------------------------------
__global__ __launch_bounds__(256) void attn_gp_kernel(
    const float* __restrict__ q, const float* __restrict__ k,
    float* __restrict__ gp, int S, int E)
{
    __shared__ float qs[196 * HHD];
    __shared__ float ks[196 * HHD];
    __shared__ float gps[196];

    const int be = blockIdx.x;
    const int b = be / E, e = be % E;
    const int nt = blockDim.x, tid = threadIdx.x;

    for (int t = tid; t < S * HHD; t += nt) {
        const int s = t / HHD, h = t % HHD;
        const long off = ((long)(b * S + s) * HHD + h) * E + e;
        qs[s * HHD + h] = q[off];
        ks[s * HHD + h] = k[off];
    }
    for (int t = tid; t < S; t += nt) gps[t] = 0.f;
    __syncthreads();

    const float scale = 0.5773502691896258f;  // 1/sqrt(H)
    for (int i = tid; i < S; i += nt) {
        const float q0 = qs[i * 3], q1 = qs[i * 3 + 1], q2 = qs[i * 3 + 2];
        float m = -3.4e38f, ssum = 0.f;
        for (int j = 0; j < S; ++j) {               // online softmax stats
            const float d = (q0 * ks[j * 3] + q1 * ks[j * 3 + 1] + q2 * ks[j * 3 + 2]) * scale;
            if (d > m) { ssum = ssum * __expf(m - d) + 1.f; m = d; }
            else       { ssum += __expf(d - m); }
        }
        const float inv = 1.f / ssum;
        for (int j = 0; j < S; ++j) {               // accumulate column sums
            const float d = (q0 * ks[j * 3] + q1 * ks[j * 3 + 1] + q2 * ks[j * 3 + 2]) * scale;
            atomicAdd(&gps[j], __expf(d - m) * inv);   // ds_add_f32
        }
    }
    __syncthreads();
    for (int t = tid; t < S; t += nt) gp[(long)be * S + t] = gps[t];
}

// ---------------------------------------------------------------------------
// Top-3 per (b,e). Strict '>' scan keeps first occurrence (JAX tie rule).
// ---------------------------------------------------------------------------
__global__ void topk_kernel(const float* __restrict__ gp, float* __restrict__ vals,
                            int* __restrict__ idx, int BE, int S)
{
    const int t = blockIdx.x * blockDim.x + threadIdx.x;
    if (t >= BE) return;
    const float* g = gp + (long)t * S;
    int s0 = -1, s1 = -1, s2 = -1;
    for (int kk = 0; kk < KK; ++kk) {
        float best = -3.4e38f; int bi = 0;
        for (int j = 0; j < S; ++j) {
            if (j == s0 || j == s1 || j == s2) continue;
            const float v = g[j];
            if (v > best) { best = v; bi = j; }
        }
        vals[t * KK + kk] = best;
        idx[t * KK + kk]  = bi;
        if (kk == 0) s0 = bi; else if (kk == 1) s1 = bi; else s2 = bi;
    }
}

// ---------------------------------------------------------------------------
// inp[b,e,kk*D+d] = x[b, idx[b,e,kk], d] * vals[b,e,kk]
// ---------------------------------------------------------------------------
__global__ void gather_kernel(const float* __restrict__ x, const float* __restrict__ vals,
                              const int* __restrict__ idx, float* __restrict__ inp,
                              int S, int D)
{
    const long tid = (long)blockIdx.x * blockDim.x + threadIdx.x;
    const long total = (long)BB * EE * KK * D;
    if (tid >= total) return;
    const int d = (int)(tid % D);
    long r = tid / D;
    const int kk = (int)(r % KK); r /= KK;
    const int e = (int)(r % EE);
    const int b = (int)(r / EE);
    const int be = (b * EE + e) * KK + kk;
    const int token = idx[be];
    inp[tid] = x[((long)b * S + token) * D + d] * vals[be];
}

// ---------------------------------------------------------------------------
// Host-side orchestration
// ---------------------------------------------------------------------------
struct MoeP { const float *Wq,*bq,*Wk,*bk,*W1,*b1,*W2,*b2,*W3,*b3; };

static MoeP get_moe(void* const* d_in, int base) {
    MoeP p;
    p.Wq = (const float*)d_in[base + 0]; p.bq = (const float*)d_in[base + 1];
    p.Wk = (const float*)d_in[base + 2]; p.bk = (const float*)d_in[base + 3];
    p.W1 = (const float*)d_in[base + 4]; p.b1 = (const float*)d_in[base + 5];
    p.W2 = (const float*)d_in[base + 6]; p.b2 = (const float*)d_in[base + 7];
    p.W3 = (const float*)d_in[base + 8]; p.b3 = (const float*)d_in[base + 9];
    return p;
}

static void moe_layer(hipStream_t stream, const float* x, int S, int D, const MoeP& p,
                      float* q, float* k, float* gp, float* vals, int* idx,
                      float* inp, float* h1, float* h2, float* out, int actLast)
{
    const int M  = BB * S;
    const int KD = KK * D;
    const dim3 blk(128);

    // q/k projections: (M x D) @ (D x 60) + b
    wmma_linear_kernel<HE><<<dim3(1, (M + 15) / 16, 1), blk, 0, stream>>>(
        x, 0, D, p.Wq, 0, p.bq, 0, q, 0, HE, M, HE, D, 0);
    wmma_linear_kernel<HE><<<dim3(1, (M + 15) / 16, 1), blk, 0, stream>>>(
        x, 0, D, p.Wk, 0, p.bk, 0, k, 0, HE, M, HE, D, 0);

    attn_gp_kernel<<<dim3(BB * EE), dim3(256), 0, stream>>>(q, k, gp, S, EE);
    topk_kernel<<<dim3((BB * EE + 127) / 128), dim3(128), 0, stream>>>(gp, vals, idx, BB * EE, S);

    const long total = (long)BB * EE * KK * D;
    gather_kernel<<<dim3((unsigned)((total + 255) / 256)), dim3(256), 0, stream>>>(
        x, vals, idx, inp, S, D);

    // Per-expert FFN (batched over grid.z = E)
    // h1 = relu(inp_e[B,KD] @ W1_e[KD,128] + b1_e)
    wmma_linear_kernel<128><<<dim3(2, 4, EE), blk, 0, stream>>>(
        inp, KD, EE * KD, p.W1, (long)KD * 128, p.b1, 128,
        h1, 128, EE * 128, BB, 128, KD, 1);
    // h2 = relu(h1_e @ W2_e + b2_e)
    wmma_linear_kernel<128><<<dim3(2, 4, EE), blk, 0, stream>>>(
        h1, 128, EE * 128, p.W2, 128 * 128, p.b2, 128,
        h2, 128, EE * 128, BB, 128, 128, 1);
    // out = act(h2_e @ W3_e + b3_e)
    wmma_linear_kernel<128><<<dim3(2, 4, EE), blk, 0, stream>>>(
        h2, 128, EE * 128, p.W3, 128 * 128, p.b3, 128,
        out, 128, EE * 128, BB, 128, 128, actLast);
}

extern "C" void kernel_launch(void* const* d_in, const int* in_sizes, int n_in,
                              void* d_out, int out_size, void* d_ws, size_t ws_size,
                              hipStream_t stream)
{
    (void)in_sizes; (void)n_in; (void)out_size; (void)ws_size;

    const float* x = (const float*)d_in[0];
    const MoeP m1 = get_moe(d_in, 1);
    const MoeP m2 = get_moe(d_in, 11);
    const MoeP m3 = get_moe(d_in, 21);
    const float* fc1W = (const float*)d_in[31]; const float* fc1b = (const float*)d_in[32];
    const float* fc2W = (const float*)d_in[33]; const float* fc2b = (const float*)d_in[34];
    const float* fc3W = (const float*)d_in[35]; const float* fc3b = (const float*)d_in[36];
    const float* l1W  = (const float*)d_in[37]; const float* l1b  = (const float*)d_in[38];
    const float* l2W  = (const float*)d_in[39]; const float* l2b  = (const float*)d_in[40];

    // Workspace carve-out (floats), sized for the largest (moe1) shapes.
    float* ws   = (float*)d_ws;
    float* q    = ws;                         // 64*196*60 = 752640
    float* k    = q    + 752640;              // 752640
    float* gp   = k    + 752640;              // 64*20*196 = 250880
    float* vals = gp   + 250880;              // 64*20*3 = 3840
    int*   idx  = (int*)(vals + 3840);        // 3840 ints
    float* inp  = (float*)(idx + 3840);       // 64*20*3*768 = 2949120
    float* h1   = inp  + 2949120;             // 64*20*128 = 163840
    float* h2   = h1   + 163840;
    float* moeo = h2   + 163840;              // (B, 2560) contiguous
    float* fco  = moeo + 163840;              // (B, 2560)
    float* l1o  = fco  + 163840;              // (B, 128)

    const dim3 blk(128);

    // MoE-1 on x (S=196, D=768), sigmoid fused into last expert GEMM
    moe_layer(stream, x, 196, 768, m1, q, k, gp, vals, idx, inp, h1, h2, moeo, 2);
    // fc1: relu((64,2560) @ (2560,2560) + b)
    wmma_linear_kernel<2560><<<dim3(40, 4, 1), blk, 0, stream>>>(
        moeo, 0, 2560, fc1W, 0, fc1b, 0, fco, 0, 2560, BB, 2560, 2560, 1);
    // MoE-2 (S=20, D=128), relu fused
    moe_layer(stream, fco, 20, 128, m2, q, k, gp, vals, idx, inp, h1, h2, moeo, 1);
    // fc2
    wmma_linear_kernel<2560><<<dim3(40, 4, 1), blk, 0, stream>>>(
        moeo, 0, 2560, fc2W, 0, fc2b, 0, fco, 0, 2560, BB, 2560, 2560, 1);
    // MoE-3 (S=20, D=128), sigmoid fused
    moe_layer(stream, fco, 20, 128, m3, q, k, gp, vals, idx, inp, h1, h2, moeo, 2);
    // fc3
    wmma_linear_kernel<2560><<<dim3(40, 4, 1), blk, 0, stream>>>(
        moeo, 0, 2560, fc3W, 0, fc3b, 0, fco, 0, 2560, BB, 2560, 2560, 1);
    // last1: (64,2560) @ (2560,128)
    wmma_linear_kernel<128><<<dim3(2, 4, 1), blk, 0, stream>>>(
        fco, 0, 2560, l1W, 0, l1b, 0, l1o, 0, 128, BB, 128, 2560, 0);
    // last2: (64,128) @ (128,10) -> d_out
    wmma_linear_kernel<10><<<dim3(1, 4, 1), blk, 0, stream>>>(
        l1o, 0, 128, l2W, 0, l2b, 0, (float*)d_out, 0, 10, BB, 10, 128, 0);
}